// Encoding_86526411145824
// MI455X (gfx1250) — compile-verified
//
#include <hip/hip_runtime.h>

// Problem constants (from reference)
#define Bn 16
#define Dn 256
#define Kn 64
#define Nn 9216              // 96*96
#define NT 64                // n-positions per LDS tile
#define CHUNK 512            // n-positions per block
#define ITERS (CHUNK / NT)   // 8
#define NBLK (Nn / CHUNK)    // 18

// Padded LDS strides (floats)
#define DS_T 68              // XT[Dn][DS_T]  (TDM pads 64-dword lines by 4 dwords)
#define DS_C 257             // Cs[Kn][DS_C]
#define DS_A 68              // As[NT][DS_A]

typedef float v2f __attribute__((ext_vector_type(2)));
typedef float v8f __attribute__((ext_vector_type(8)));
typedef unsigned int v4u __attribute__((ext_vector_type(4)));
typedef int v8i __attribute__((ext_vector_type(8)));
typedef int v4i __attribute__((ext_vector_type(4)));

#define HAVE_TDM __has_builtin(__builtin_amdgcn_tensor_load_to_lds)

__global__ void enc_zero(float* __restrict__ out, float* __restrict__ asum) {
    int i = blockIdx.x * blockDim.x + threadIdx.x;
    if (i < Bn * Kn * Dn) out[i] = 0.0f;
    if (i < Bn * Kn) asum[i] = 0.0f;
}

#if HAVE_TDM
// One TDM descriptor: load tile [NT n-contig x Dn d-rows] (fp32) starting at
// element offset gelem_off of X (row stride Nn elements) into LDS at lds_off,
// padding each 64-dword line by 4 dwords (-> 68-float LDS row stride).
__device__ __forceinline__ void tdm_load_tile(const float* X, int gelem_off,
                                              unsigned lds_off) {
    unsigned long long gaddr =
        (unsigned long long)(uintptr_t)X + (unsigned long long)gelem_off * 4ull;
    v4u g0;
    g0.x = 1u;                                   // count=1 (valid), load, normal
    g0.y = lds_off;                              // LDS byte address
    g0.z = (unsigned)(gaddr & 0xffffffffu);      // global_addr[31:0]
    g0.w = (unsigned)((gaddr >> 32) & 0x1ffffffu) | (2u << 30);  // [56:32] | type=2

    v8i g1;
    g1[0] = (2 << 16)        // data_size = 4 bytes
          | (1 << 20)        // pad_enable
          | (5 << 22)        // pad_interval: 64 dwords
          | (3 << 25);       // pad_amount: 4 dwords
    g1[1] = (Nn & 0xffff) << 16;                   // tensor_dim0[15:0]
    g1[2] = (Nn >> 16) | ((Dn & 0xffff) << 16);    // tensor_dim0[31:16] | dim1[15:0]
    g1[3] = (NT << 16);                            // dim1[31:16]=0 | tile_dim0=NT
    g1[4] = Dn;                                    // tile_dim1=Dn | tile_dim2=0
    g1[5] = Nn;                                    // tensor_dim0_stride[31:0]
    g1[6] = 0;                                     // stride0[47:32] | stride1[15:0]
    g1[7] = 0;                                     // tensor_dim1_stride[47:16]

    v4i g2 = (v4i){0, 0, 0, 0};
    v4i g3 = (v4i){0, 0, 0, 0};
    v8i g4 = (v8i){0, 0, 0, 0, 0, 0, 0, 0};
    __builtin_amdgcn_tensor_load_to_lds(g0, g1, g2, g3, g4, 0);
}
#endif

__global__ __launch_bounds__(256) void enc_main(const float* __restrict__ X,
                                                const float* __restrict__ Cw,
                                                const float* __restrict__ scale,
                                                float* __restrict__ out,
                                                float* __restrict__ asum) {
    extern __shared__ float smem[];
    float* XT0 = smem;                     // Dn * DS_T  (X^T tile, buffer 0)
    float* XT1 = XT0 + Dn * DS_T;          // Dn * DS_T  (buffer 1)
    float* Cs  = XT1 + Dn * DS_T;          // Kn * DS_C
    float* As  = Cs + Kn * DS_C;           // NT * DS_A
    float* x2s = As + NT * DS_A;           // NT
    float* c2s = x2s + NT;                 // Kn
    float* scs = c2s + Kn;                 // Kn

    const int tid  = threadIdx.x;
    const int lane = tid & 31;
    const int w    = tid >> 5;             // wave id, 8 waves
    const int b    = blockIdx.x / NBLK;
    const int chunk = blockIdx.x % NBLK;
    const int xbase = b * Dn * Nn;
    const int nbase = chunk * CHUNK;

#if HAVE_TDM
    // Prologue: wave 0 kicks the DMA for tile 0 immediately.
    if (tid < 32) {
        tdm_load_tile(X, xbase + nbase, (unsigned)(uintptr_t)(void*)XT0);
    }
#endif

    // Stage codewords once per block (coalesced)
    for (int idx = tid; idx < Kn * Dn; idx += 256) {
        int k = idx >> 8, d = idx & 255;
        Cs[k * DS_C + d] = Cw[idx];
    }
    if (tid < Kn) scs[tid] = scale[tid];
    __syncthreads();
    if (tid < Kn) {
        float s = 0.0f;
        for (int d = 0; d < Dn; ++d) { float c = Cs[tid * DS_C + d]; s += c * c; }
        c2s[tid] = s;
    }

    // Persistent GEMM2 accumulators: wave w owns k-tile (w&3), d-tiles ((w>>2)*8 .. +7)
    const int kt2 = w & 3;
    const int dtb = (w >> 2) * 8;
    v8f eacc[8];
    #pragma unroll
    for (int j = 0; j < 8; ++j) eacc[j] = (v8f){0,0,0,0,0,0,0,0};
    float asum_reg = 0.0f;

    // GEMM1 tile assignment: wave w owns n-tile (w&3), k-tiles ((w>>2)*2, +1)
    const int nt1 = w & 3;
    const int kt1 = (w >> 2) * 2;

    for (int it = 0; it < ITERS; ++it) {
        float* XT = (it & 1) ? XT1 : XT0;
        __syncthreads();  // all waves done with the spare buffer + previous As

#if HAVE_TDM
        if (tid < 32) {
            float* XTnext = (it & 1) ? XT0 : XT1;
            if (it + 1 < ITERS) {
                tdm_load_tile(X, xbase + nbase + (it + 1) * NT,
                              (unsigned)(uintptr_t)(void*)XTnext);
                __builtin_amdgcn_s_wait_tensorcnt(1);  // oldest (current tile) done
            } else {
                __builtin_amdgcn_s_wait_tensorcnt(0);
            }
        }
#else
        // Fallback: manual staging into the same transposed/padded layout
        const int n0 = nbase + it * NT;
        for (int idx = tid; idx < NT * Dn; idx += 256) {
            int n = idx & (NT - 1), d = idx >> 6;
            XT[d * DS_T + n] = X[xbase + d * Nn + n0 + n];
        }
#endif
        __syncthreads();  // tile data visible to all waves

        if (tid < NT) {
            float s = 0.0f;
            for (int d = 0; d < Dn; ++d) { float x = XT[d * DS_T + tid]; s += x * x; }
            x2s[tid] = s;
        }

        // ---- GEMM1: xc[nt1 stripe][64 k] = Xtile * Cw^T via f32 WMMA ----
        v8f xc0 = (v8f){0,0,0,0,0,0,0,0};
        v8f xc1 = (v8f){0,0,0,0,0,0,0,0};
        {
            const int an  = nt1 * 16 + (lane & 15);
            const int dhi = (lane >> 4) * 2;
            const int bk0 = (kt1 + 0) * 16 + (lane & 15);
            const int bk1 = (kt1 + 1) * 16 + (lane & 15);
            #pragma unroll 8
            for (int d0 = 0; d0 < Dn; d0 += 4) {
                v2f a, b0, b1;
                a.x  = XT[(d0 + dhi) * DS_T + an];
                a.y  = XT[(d0 + dhi + 1) * DS_T + an];
                b0.x = Cs[bk0 * DS_C + d0 + dhi];
                b0.y = Cs[bk0 * DS_C + d0 + dhi + 1];
                b1.x = Cs[bk1 * DS_C + d0 + dhi];
                b1.y = Cs[bk1 * DS_C + d0 + dhi + 1];
                xc0 = __builtin_amdgcn_wmma_f32_16x16x4_f32(false, a, false, b0,
                                                            (short)0, xc0, false, false);
                xc1 = __builtin_amdgcn_wmma_f32_16x16x4_f32(false, a, false, b1,
                                                            (short)0, xc1, false, false);
            }
        }
        // Spill xc tiles to LDS As[n][k]
        {
            const int nrow = nt1 * 16 + 8 * (lane >> 4);
            const int kc   = lane & 15;
            #pragma unroll
            for (int r = 0; r < 8; ++r) {
                As[(nrow + r) * DS_A + (kt1 + 0) * 16 + kc] = xc0[r];
                As[(nrow + r) * DS_A + (kt1 + 1) * 16 + kc] = xc1[r];
            }
        }
        __syncthreads();

        // ---- Softmax over K per position (waves 0-1; whole-wave branches) ----
        if (tid < NT) {
            const float x2v = x2s[tid];
            float m = -3.4e38f;
            for (int k = 0; k < Kn; ++k) {
                float sl = scs[k] * (x2v - 2.0f * As[tid * DS_A + k] + c2s[k]);
                As[tid * DS_A + k] = sl;
                m = fmaxf(m, sl);
            }
            float s = 0.0f;
            for (int k = 0; k < Kn; ++k) {
                float e = __expf(As[tid * DS_A + k] - m);
                As[tid * DS_A + k] = e;
                s += e;
            }
            float inv = 1.0f / s;
            for (int k = 0; k < Kn; ++k) As[tid * DS_A + k] *= inv;
        }
        __syncthreads();

        // Column sums of A (for the -sum(A)*C term)
        if (tid < Kn) {
            float s = 0.0f;
            for (int n = 0; n < NT; ++n) s += As[n * DS_A + tid];
            asum_reg += s;
        }

        // ---- GEMM2: E += A^T * Xtile via f32 WMMA (persistent accumulators) ----
        {
            const int ak  = kt2 * 16 + (lane & 15);
            const int nhi = (lane >> 4) * 2;
            #pragma unroll 4
            for (int nn = 0; nn < NT; nn += 4) {
                v2f a;
                a.x = As[(nn + nhi) * DS_A + ak];
                a.y = As[(nn + nhi + 1) * DS_A + ak];
                #pragma unroll
                for (int j = 0; j < 8; ++j) {
                    const int dcol = (dtb + j) * 16 + (lane & 15);
                    v2f bb;
                    bb.x = XT[dcol * DS_T + nn + nhi];      // adjacent floats ->
                    bb.y = XT[dcol * DS_T + nn + nhi + 1];  // single ds_load_b64
                    eacc[j] = __builtin_amdgcn_wmma_f32_16x16x4_f32(false, a, false, bb,
                                                                    (short)0, eacc[j],
                                                                    false, false);
                }
            }
        }
    }

    // Block epilogue: atomically fold partial E and asum into global buffers
    {
        const int krow = kt2 * 16 + 8 * (lane >> 4);
        const int dcol = lane & 15;
        #pragma unroll
        for (int j = 0; j < 8; ++j) {
            #pragma unroll
            for (int r = 0; r < 8; ++r) {
                atomicAdd(&out[(b * Kn + krow + r) * Dn + (dtb + j) * 16 + dcol], eacc[j][r]);
            }
        }
    }
    if (tid < Kn) atomicAdd(&asum[b * Kn + tid], asum_reg);
}

__global__ void enc_final(float* __restrict__ out, const float* __restrict__ asum,
                          const float* __restrict__ Cw) {
    int i = blockIdx.x * 256 + threadIdx.x;   // i < Bn*Kn*Dn
    int d = i & 255;
    int k = (i >> 8) & 63;
    out[i] -= asum[i >> 8] * Cw[k * Dn + d];
}

extern "C" void kernel_launch(void* const* d_in, const int* in_sizes, int n_in,
                              void* d_out, int out_size, void* d_ws, size_t ws_size,
                              hipStream_t stream) {
    const float* X  = (const float*)d_in[0];   // [B, D, H, W] fp32
    const float* Cw = (const float*)d_in[1];   // [K, D] fp32
    const float* sc = (const float*)d_in[2];   // [K] fp32
    float* out  = (float*)d_out;               // [B, K, D] fp32
    float* asum = (float*)d_ws;                // [B, K] fp32 scratch

    enc_zero<<<(Bn * Kn * Dn + 255) / 256, 256, 0, stream>>>(out, asum);

    const size_t shmem = (size_t)(2 * Dn * DS_T + Kn * DS_C + NT * DS_A + NT + Kn + Kn)
                         * sizeof(float);
    enc_main<<<Bn * NBLK, 256, shmem, stream>>>(X, Cw, sc, out, asum);

    enc_final<<<(Bn * Kn * Dn) / 256, 256, 0, stream>>>(out, asum, Cw);
}